// GraphReadout_52733608460567
// MI455X (gfx1250) — compile-verified
//
#include <hip/hip_runtime.h>
#include <math.h>

#define FEAT_D     256
#define NUM_GRAPHS 1024
#define PIPE       4                 // async tiles in flight
#define TILE_ROWS  4                 // 256 threads x 16B = 4 rows per tile
#define STAGE_FLOATS (256 * 4)       // floats per stage buffer (4 KB)

typedef float v4f __attribute__((ext_vector_type(4)));

__device__ __forceinline__ int lower_bound_i(const int* __restrict__ batch, int n, int key) {
    int lo = 0, hi = n;
    while (lo < hi) {
        int mid = (lo + hi) >> 1;
        if (batch[mid] < key) lo = mid + 1; else hi = mid;
    }
    return lo;
}

__global__ __launch_bounds__(256)
void GraphReadout_52733608460567_kernel(const float* __restrict__ x,
                                        const int*   __restrict__ batch,
                                        float*       __restrict__ out,
                                        int n) {
    const int g   = blockIdx.x;      // one block per segment (batch is sorted)
    const int tid = threadIdx.x;

    __shared__ int   s_bounds[2];
    __shared__ float s_stage[PIPE][STAGE_FLOATS];  // 16 KB async staging ring
    __shared__ v4f   s_sum[256];
    __shared__ v4f   s_max[256];

    if (tid == 0) {
        s_bounds[0] = lower_bound_i(batch, n, g);
        s_bounds[1] = lower_bound_i(batch, n, g + 1);
    }
    __syncthreads();
    const int start = s_bounds[0];
    const int end   = s_bounds[1];
    const int rows  = end - start;

    const int c4 = (tid & 63) << 2;  // this thread's column quad
    const int rg = tid >> 6;         // row phase 0..3 within a tile

    v4f sum = (v4f){0.f, 0.f, 0.f, 0.f};
    v4f mx  = (v4f){-INFINITY, -INFINITY, -INFINITY, -INFINITY};

    if (rows > 0) {
        const int ntiles = (rows + TILE_ROWS - 1) / TILE_ROWS;
        // low 32 bits of the generic pointer = LDS byte offset (aperture layout)
        const unsigned lds0 = (unsigned)(uintptr_t)&s_stage[0][tid * 4];

        // async-stage tile t: 16 B per lane, rows clamped into the segment so
        // over-issued pipeline tiles always read valid memory (data masked later)
        auto stage = [&](int t) {
            int r = start + t * TILE_ROWS + rg;
            r = (r < end) ? r : (end - 1);
            const float* gp = x + (size_t)r * FEAT_D + c4;
            const unsigned dst =
                lds0 + (unsigned)((t & (PIPE - 1)) * (STAGE_FLOATS * sizeof(float)));
            asm volatile("global_load_async_to_lds_b128 %0, %1, off"
                         :: "v"(dst), "v"(gp) : "memory");
        };

#pragma unroll
        for (int t = 0; t < PIPE; ++t) stage(t);   // prologue: fill pipeline

        for (int i = 0; i < ntiles; ++i) {
            // oldest of the PIPE outstanding tiles (tile i) has landed in LDS
            asm volatile("s_wait_asynccnt 0x3" ::: "memory");
            const v4f v = *(const v4f*)&s_stage[i & (PIPE - 1)][tid * 4];
            if (start + i * TILE_ROWS + rg < end) {
                sum.x += v.x; sum.y += v.y; sum.z += v.z; sum.w += v.w;
                mx.x = fmaxf(mx.x, v.x); mx.y = fmaxf(mx.y, v.y);
                mx.z = fmaxf(mx.z, v.z); mx.w = fmaxf(mx.w, v.w);
            }
            // ensure our ds_load of this buffer retired before the async engine
            // overwrites it with tile i+PIPE
            asm volatile("s_wait_dscnt 0x0" ::: "memory");
            stage(i + PIPE);
        }
        asm volatile("s_wait_asynccnt 0x0" ::: "memory");  // drain ring
    }

    s_sum[tid] = sum;
    s_max[tid] = mx;
    __syncthreads();

    // threads with rg==0 (two wave32s) fold the 4 row phases and emit results
    if (rg == 0) {
#pragma unroll
        for (int k = 1; k < 4; ++k) {
            const v4f a = s_sum[tid + (k << 6)];
            const v4f b = s_max[tid + (k << 6)];
            sum.x += a.x; sum.y += a.y; sum.z += a.z; sum.w += a.w;
            mx.x = fmaxf(mx.x, b.x); mx.y = fmaxf(mx.y, b.y);
            mx.z = fmaxf(mx.z, b.z); mx.w = fmaxf(mx.w, b.w);
        }
        const float cnt = (float)rows;
        const float inv = 1.0f / fmaxf(cnt, 1.0f);
        const v4f mean = (v4f){sum.x * inv, sum.y * inv, sum.z * inv, sum.w * inv};

        float* o = out + (size_t)g * (3 * FEAT_D);
        *(v4f*)(o + c4)              = mean;  // mean_pool
        *(v4f*)(o + FEAT_D + c4)     = mx;    // max_pool
        *(v4f*)(o + 2 * FEAT_D + c4) = sum;   // sum_pool
    }
}

extern "C" void kernel_launch(void* const* d_in, const int* in_sizes, int n_in,
                              void* d_out, int out_size, void* d_ws, size_t ws_size,
                              hipStream_t stream) {
    const float* x     = (const float*)d_in[0];
    const int*   batch = (const int*)d_in[1];
    float*       out   = (float*)d_out;
    const int n = in_sizes[1];  // number of nodes (batch length)

    GraphReadout_52733608460567_kernel<<<NUM_GRAPHS, 256, 0, stream>>>(x, batch, out, n);
}